// MlpwithSOMModule_32452772889032
// MI455X (gfx1250) — compile-verified
//
#include <hip/hip_runtime.h>

typedef float v2f __attribute__((ext_vector_type(2)));
typedef float v8f __attribute__((ext_vector_type(8)));

#define BCN 512
#define LR  128
#define DD  768
#define HH  768
#define SA  36    // LDS stride (floats) for 128x32 A/ctx/ent tiles
#define SW  136   // LDS stride (floats) for 32x128 W1 tile
#define NT1 24    // 768/32 K-tiles, phase 1
#define NT2 48    // 1536/32 K-tiles, phase 2

// ---- CDNA5 async global->LDS copy (ASYNCcnt), inline asm per CDNA5 ISA ----
__device__ __forceinline__ void async_ld_b128(unsigned lds_byte_off, const void* gptr) {
    asm volatile("global_load_async_to_lds_b128 %0, %1, off"
                 :: "v"(lds_byte_off), "v"(gptr) : "memory");
}
__device__ __forceinline__ void wait_async_le8() {
    asm volatile("s_wait_asynccnt 0x8" ::: "memory");
}
__device__ __forceinline__ void wait_async_0() {
    asm volatile("s_wait_asynccnt 0x0" ::: "memory");
}
__device__ __forceinline__ unsigned lds_off(const void* p) {
    return (unsigned)(unsigned long long)p;   // low 32 bits of generic ptr = LDS byte offset
}

__device__ __forceinline__ float block_sum_256(float v, float* s_red, int tid) {
    #pragma unroll
    for (int off = 16; off >= 1; off >>= 1) v += __shfl_xor(v, off, 32);
    const int wave = tid >> 5, lane = tid & 31;
    if (lane == 0) s_red[wave] = v;
    __syncthreads();
    float r = (tid < 8) ? s_red[tid] : 0.0f;
    if (wave == 0) {
        #pragma unroll
        for (int off = 4; off >= 1; off >>= 1) r += __shfl_xor(r, off, 32);
    }
    __syncthreads();
    return r; // valid in thread 0
}

__global__ __launch_bounds__(256, 2)
void mlp_som_fused(const float* __restrict__ input,
                   const float* __restrict__ W1,
                   const float* __restrict__ b1,
                   const float* __restrict__ W2,
                   const float* __restrict__ b2,
                   float* __restrict__ out) {
    __shared__ float ldsA[2][LR * SA];   // 2 x 18432 B : ctx tile / mlp_in tile (ping-pong)
    __shared__ float ldsB[2][LR * SA];   // 2 x 18432 B : ent tile (p1) / W1 tile 32xSW (p2)
    __shared__ int   s_arg[LR];
    __shared__ float s_max[LR];
    __shared__ float s_mask[LR];
    __shared__ float s_score[LR];
    __shared__ float s_b1[HH];
    __shared__ float s_w2[HH];
    __shared__ float s_red[8];

    const int bc   = blockIdx.x;
    const int tid  = threadIdx.x;
    const int wave = tid >> 5;
    const int lane = tid & 31;
    const int lm   = lane & 15;
    const int hi   = lane >> 4;
    const int m0   = wave * 16;

    const float* ctx = input + (size_t)bc * 2 * LR * DD;
    const float* ent = ctx + (size_t)LR * DD;
    const float  b2v = b2[0];

    // per-thread chunk coordinates for the 128x32 tiles (4 x b128 per thread)
    int rA[4], cA[4];
    #pragma unroll
    for (int i = 0; i < 4; ++i) {
        const int v = tid + i * 256;
        rA[i] = v >> 3;
        cA[i] = (v & 7) << 2;
    }

    // ---------------- Phase 1 prologue: async-issue tile 0 ----------------
    #pragma unroll
    for (int i = 0; i < 4; ++i) {
        async_ld_b128(lds_off(&ldsA[0][rA[i] * SA + cA[i]]), ctx + (size_t)rA[i] * DD + cA[i]);
        async_ld_b128(lds_off(&ldsB[0][rA[i] * SA + cA[i]]), ent + (size_t)rA[i] * DD + cA[i]);
    }

    // preload bias / W2 column / mask (overlaps with in-flight async tile 0)
    for (int i = tid; i < HH; i += 256) { s_b1[i] = b1[i]; s_w2[i] = W2[i]; }
    for (int r = tid; r < LR; r += 256) s_mask[r] = (ctx[(size_t)r * DD] != 0.0f) ? 1.0f : 0.0f;

    // ---------------- Phase 1: sim = ctx @ ent^T, row max/argmax ----------------
    v8f acc[8];
    #pragma unroll
    for (int nn = 0; nn < 8; ++nn)
        #pragma unroll
        for (int j = 0; j < 8; ++j) acc[nn][j] = 0.0f;

    for (int kt = 0; kt < NT1; ++kt) {
        const int cur = kt & 1;
        if (kt + 1 < NT1) {
            const int nxt = cur ^ 1;
            const int k1  = (kt + 1) * 32;
            #pragma unroll
            for (int i = 0; i < 4; ++i) {
                async_ld_b128(lds_off(&ldsA[nxt][rA[i] * SA + cA[i]]),
                              ctx + (size_t)rA[i] * DD + k1 + cA[i]);
                async_ld_b128(lds_off(&ldsB[nxt][rA[i] * SA + cA[i]]),
                              ent + (size_t)rA[i] * DD + k1 + cA[i]);
            }
            wait_async_le8();   // tile kt resident (async ops complete in order)
        } else {
            wait_async_0();
        }
        __syncthreads();        // all waves' tile-kt data visible

        const float* __restrict__ tA = ldsA[cur];
        const float* __restrict__ tB = ldsB[cur];
        #pragma unroll
        for (int kk = 0; kk < 32; kk += 4) {
            v2f a;
            a.x = tA[(m0 + lm) * SA + kk + 2 * hi];
            a.y = tA[(m0 + lm) * SA + kk + 2 * hi + 1];
            #pragma unroll
            for (int nn = 0; nn < 8; ++nn) {
                v2f b;
                b.x = tB[(nn * 16 + lm) * SA + kk + 2 * hi];
                b.y = tB[(nn * 16 + lm) * SA + kk + 2 * hi + 1];
                acc[nn] = __builtin_amdgcn_wmma_f32_16x16x4_f32(
                    false, a, false, b, (short)0, acc[nn], false, false);
            }
        }
        __syncthreads();        // buffer free before tile kt+2 targets it
    }

    // register-space row max + first-occurrence argmax
    // C layout: acc[nn] vgpr j, lanes 0-15 -> row m0+j col nn*16+lm; lanes 16-31 -> row m0+j+8
    #pragma unroll
    for (int j = 0; j < 8; ++j) {
        float mv = acc[0][j];
        int   mi = lm;
        #pragma unroll
        for (int nn = 1; nn < 8; ++nn) {
            const float v = acc[nn][j];
            const int   c = nn * 16 + lm;
            if (v > mv) { mv = v; mi = c; }
        }
        #pragma unroll
        for (int off = 1; off < 16; off <<= 1) {
            const float ov = __shfl_xor(mv, off, 32);
            const int   oi = __shfl_xor(mi, off, 32);
            if (ov > mv || (ov == mv && oi < mi)) { mv = ov; mi = oi; }
        }
        if (lm == 0) {
            const int row = m0 + j + 8 * hi;
            s_max[row] = mv;
            s_arg[row] = mi;
        }
    }
    __syncthreads();

    // dot_scores[bc] = sum_r max_r * mask_r
    {
        const float dp = (tid < LR) ? s_max[tid] * s_mask[tid] : 0.0f;
        const float ds = block_sum_256(dp, s_red, tid);
        if (tid == 0) out[BCN + bc] = ds;
    }

    // per-thread chunk coordinates for the 32x128 W1 tiles (4 x b128 per thread)
    int rW[4], cW[4];
    #pragma unroll
    for (int i = 0; i < 4; ++i) {
        const int v = tid + i * 256;
        rW[i] = v >> 5;          // 0..31
        cW[i] = (v & 31) << 2;   // 0..124
    }

    // ---------------- Phase 2: h = relu([ctx|gather] @ W1 + b1); score = h @ W2 ----------------
    float rs[8];
    #pragma unroll
    for (int j = 0; j < 8; ++j) rs[j] = 0.0f;

    for (int nb = 0; nb < 6; ++nb) {
        const int n0 = nb * 128;
        #pragma unroll
        for (int nn = 0; nn < 8; ++nn)
            #pragma unroll
            for (int j = 0; j < 8; ++j) acc[nn][j] = 0.0f;

        // prologue: async-issue k-tile 0 for this N-block
        #pragma unroll
        for (int i = 0; i < 4; ++i) {
            async_ld_b128(lds_off(&ldsA[0][rA[i] * SA + cA[i]]),
                          ctx + (size_t)rA[i] * DD + cA[i]);
            async_ld_b128(lds_off(&ldsB[0][rW[i] * SW + cW[i]]),
                          W1 + (size_t)rW[i] * HH + n0 + cW[i]);
        }

        for (int kt = 0; kt < NT2; ++kt) {
            const int cur = kt & 1;
            if (kt + 1 < NT2) {
                const int nxt = cur ^ 1;
                const int k1  = (kt + 1) * 32;
                const bool second = (k1 >= DD);
                const int  kb     = second ? (k1 - DD) : k1;
                #pragma unroll
                for (int i = 0; i < 4; ++i) {
                    const float* rowp = second ? (ent + (size_t)s_arg[rA[i]] * DD + kb)
                                               : (ctx + (size_t)rA[i] * DD + kb);
                    async_ld_b128(lds_off(&ldsA[nxt][rA[i] * SA + cA[i]]), rowp + cA[i]);
                    async_ld_b128(lds_off(&ldsB[nxt][rW[i] * SW + cW[i]]),
                                  W1 + (size_t)(k1 + rW[i]) * HH + n0 + cW[i]);
                }
                wait_async_le8();
            } else {
                wait_async_0();
            }
            __syncthreads();

            const float* __restrict__ tA = ldsA[cur];
            const float* __restrict__ tW = ldsB[cur];
            #pragma unroll
            for (int kk = 0; kk < 32; kk += 4) {
                v2f a;
                a.x = tA[(m0 + lm) * SA + kk + 2 * hi];
                a.y = tA[(m0 + lm) * SA + kk + 2 * hi + 1];
                #pragma unroll
                for (int nn = 0; nn < 8; ++nn) {
                    v2f b;
                    b.x = tW[(kk + 2 * hi) * SW + nn * 16 + lm];
                    b.y = tW[(kk + 2 * hi + 1) * SW + nn * 16 + lm];
                    acc[nn] = __builtin_amdgcn_wmma_f32_16x16x4_f32(
                        false, a, false, b, (short)0, acc[nn], false, false);
                }
            }
            __syncthreads();
        }

        // fused bias + ReLU + dot with W2 column
        #pragma unroll
        for (int nn = 0; nn < 8; ++nn) {
            const int col = n0 + nn * 16 + lm;
            const float w2c = s_w2[col];
            const float bc1 = s_b1[col];
            #pragma unroll
            for (int j = 0; j < 8; ++j) {
                float h = acc[nn][j] + bc1;
                h = fmaxf(h, 0.0f);
                rs[j] += h * w2c;
            }
        }
    }

    // per-row score: reduce rs[j] over the 16-lane half, add b2
    #pragma unroll
    for (int j = 0; j < 8; ++j) {
        float v = rs[j];
        #pragma unroll
        for (int off = 1; off < 16; off <<= 1) v += __shfl_xor(v, off, 32);
        if (lm == 0) s_score[m0 + j + 8 * hi] = v + b2v;
    }
    __syncthreads();

    {
        const float sp = (tid < LR) ? s_score[tid] * s_mask[tid] : 0.0f;
        const float ss = block_sum_256(sp, s_red, tid);
        if (tid == 0) out[bc] = ss;
    }
}

extern "C" void kernel_launch(void* const* d_in, const int* in_sizes, int n_in,
                              void* d_out, int out_size, void* d_ws, size_t ws_size,
                              hipStream_t stream) {
    (void)in_sizes; (void)n_in; (void)d_ws; (void)ws_size; (void)out_size;
    const float* input = (const float*)d_in[0];
    const float* W1    = (const float*)d_in[1];
    const float* b1    = (const float*)d_in[2];
    const float* W2    = (const float*)d_in[3];
    const float* b2    = (const float*)d_in[4];
    float* out = (float*)d_out;
    mlp_som_fused<<<dim3(BCN), dim3(256), 0, stream>>>(input, W1, b1, W2, b2, out);
}